// DWTLayer_36739150250538
// MI455X (gfx1250) — compile-verified
//
#include <hip/hip_runtime.h>
#include <cstdint>

typedef float v4f __attribute__((ext_vector_type(4)));

#define WAVES_PER_BLOCK 4
#define STAGES_PER_WAVE 8
#define ROWPAIR_FLOATS  1024          // 2 rows x 512 f32 = 4 KiB chunk
#define W_IN            512
#define W_OUT           256
#define SUBBAND_STRIDE  (256 * 256)   // elements between subband planes

// Haar butterfly for one output pixel, writing component `comp` of the 4 accumulators.
#define HAAR_PX(P00, P01, P10, P11, m, comp)                         \
    do {                                                             \
        float s0 = (P00) + (P01), s1 = (P10) + (P11);                \
        float d0 = (P00) - (P01), d1 = (P10) - (P11);                \
        oA[m].comp = (s0 + s1) * 0.5f;                               \
        oH[m].comp = (s0 - s1) * 0.5f;                               \
        oV[m].comp = (d0 + d1) * 0.5f;                               \
        oD[m].comp = (d0 - d1) * 0.5f;                               \
    } while (0)

__global__ __launch_bounds__(WAVES_PER_BLOCK * 32)
void dwt_haar2d_async_kernel(const float* __restrict__ x,
                             float* __restrict__ out,
                             int total_rp)
{
    // Per-wave private double buffer: 2 x 4 KiB per wave, 32 KiB per block.
    __shared__ __align__(16) float lds[WAVES_PER_BLOCK][2][ROWPAIR_FLOATS];

    const int lane = threadIdx.x & 31;
    const int warp = threadIdx.x >> 5;
    const long long rp0 =
        ((long long)blockIdx.x * WAVES_PER_BLOCK + warp) * STAGES_PER_WAVE;
    if (rp0 >= total_rp) return;

    // Low 32 bits of a generic pointer into LDS == byte offset within LDS.
    const uint32_t lbase[2] = {
        (uint32_t)(uintptr_t)&lds[warp][0][0],
        (uint32_t)(uintptr_t)&lds[warp][1][0]
    };
    const uint32_t voff = (uint32_t)lane * 16u;  // per-lane 16B slot

    // Async-DMA one 4 KiB row-pair chunk into LDS buffer `buf`:
    // 32 lanes x 8 instrs x 16B = 4096B. Tracked by ASYNCcnt.
    auto issue = [&](long long rp, int buf) {
        const uint64_t src =
            (uint64_t)(uintptr_t)(x + rp * (long long)ROWPAIR_FLOATS);
        const uint32_t la = lbase[buf] + voff;
        #pragma unroll
        for (int i = 0; i < 8; ++i) {
            asm volatile(
                "global_load_async_to_lds_b128 %0, %1, %2 offset:0"
                :
                : "v"(la + (uint32_t)(i * 512)),
                  "v"(voff + (uint32_t)(i * 512)),
                  "s"(src)
                : "memory");
        }
    };

    issue(rp0, 0);  // prologue

    #pragma unroll 1
    for (int s = 0; s < STAGES_PER_WAVE; ++s) {
        const long long rp = rp0 + s;
        if (rp >= total_rp) break;

        const bool more = (s + 1 < STAGES_PER_WAVE) && (rp + 1 < total_rp);
        if (more) {
            issue(rp + 1, (s + 1) & 1);
            // <=8 outstanding => the 8 loads of stage s have completed (in-order).
            asm volatile("s_wait_asynccnt 8" ::: "memory");
        } else {
            asm volatile("s_wait_asynccnt 0" ::: "memory");
        }

        const float* L = &lds[warp][s & 1][0];
        // This lane's 8 output pixels: px in [lane*8, lane*8+8)
        const v4f* r0 = (const v4f*)(L + lane * 16);          // row 2y
        const v4f* r1 = (const v4f*)(L + W_IN + lane * 16);   // row 2y+1

        v4f oA[2], oH[2], oV[2], oD[2];
        #pragma unroll
        for (int m = 0; m < 2; ++m) {
            const v4f a0 = r0[2 * m + 0];   // ds_load_b128
            const v4f a1 = r0[2 * m + 1];
            const v4f b0 = r1[2 * m + 0];
            const v4f b1 = r1[2 * m + 1];
            HAAR_PX(a0.x, a0.y, b0.x, b0.y, m, x);  // px 4m+0
            HAAR_PX(a0.z, a0.w, b0.z, b0.w, m, y);  // px 4m+1
            HAAR_PX(a1.x, a1.y, b1.x, b1.y, m, z);  // px 4m+2
            HAAR_PX(a1.z, a1.w, b1.z, b1.w, m, w);  // px 4m+3
        }

        // out[(img*4 + k)][y][wx], img = rp/256, y = rp%256
        const long long img = rp >> 8;
        const int       y   = (int)(rp & 255);
        const size_t ob = (((size_t)img * 4) * W_OUT + (size_t)y) * W_OUT
                        + (size_t)lane * 8;
        v4f* pA = (v4f*)(out + ob);
        v4f* pH = (v4f*)(out + ob + (size_t)SUBBAND_STRIDE);
        v4f* pV = (v4f*)(out + ob + (size_t)(2 * SUBBAND_STRIDE));
        v4f* pD = (v4f*)(out + ob + (size_t)(3 * SUBBAND_STRIDE));
        // Write-once 256 MiB stream: non-temporal 128-bit stores.
        __builtin_nontemporal_store(oA[0], pA + 0);
        __builtin_nontemporal_store(oA[1], pA + 1);
        __builtin_nontemporal_store(oH[0], pH + 0);
        __builtin_nontemporal_store(oH[1], pH + 1);
        __builtin_nontemporal_store(oV[0], pV + 0);
        __builtin_nontemporal_store(oV[1], pV + 1);
        __builtin_nontemporal_store(oD[0], pD + 0);
        __builtin_nontemporal_store(oD[1], pD + 1);
    }
}

extern "C" void kernel_launch(void* const* d_in, const int* in_sizes, int n_in,
                              void* d_out, int out_size, void* d_ws, size_t ws_size,
                              hipStream_t stream) {
    (void)n_in; (void)out_size; (void)d_ws; (void)ws_size;
    const float* x  = (const float*)d_in[0];
    float*       out = (float*)d_out;

    // Input is a flat sequence of 4 KiB row-pair chunks (2 rows x 512 f32).
    const int total_rp = in_sizes[0] / ROWPAIR_FLOATS;   // 65536 for (8,32,512,512)
    const int rp_per_block = WAVES_PER_BLOCK * STAGES_PER_WAVE;
    const int blocks = (total_rp + rp_per_block - 1) / rp_per_block;  // 2048

    dwt_haar2d_async_kernel<<<blocks, WAVES_PER_BLOCK * 32, 0, stream>>>(
        x, out, total_rp);
}